// WeightedAverageLayer_14078902796421
// MI455X (gfx1250) — compile-verified
//
#include <hip/hip_runtime.h>
#include <hip/hip_bf16.h>

typedef __attribute__((ext_vector_type(2))) float v2f;
typedef __attribute__((ext_vector_type(8))) float v8f;

#define IN_DIM 128

// ---------------- zero workspace ----------------
__global__ void wal_zero_kernel(float* __restrict__ p, size_t n) {
    size_t i = (size_t)blockIdx.x * blockDim.x + threadIdx.x;
    if (i < n) p[i] = 0.0f;
}

// ---------------- edge scatter-add (one wave32 per edge) ----------------
__global__ void wal_scatter_kernel(const float* __restrict__ x,
                                   const long long* __restrict__ ei,
                                   float* __restrict__ agg,
                                   float* __restrict__ cnt,
                                   int E) {
    int gtid = blockIdx.x * blockDim.x + threadIdx.x;
    int edge = gtid >> 5;          // wave32 per edge
    int lane = gtid & 31;
    if (edge >= E) return;

    long long s = ei[edge];              // src row
    long long t = ei[(size_t)E + edge];  // tgt row

    // gather 128 floats of x[s]: each lane takes 4 consecutive floats (float4)
    const float4* xrow = (const float4*)(x + (size_t)s * IN_DIM);
    float4 v = xrow[lane];

    float* dst = agg + (size_t)t * IN_DIM + lane * 4;
    unsafeAtomicAdd(dst + 0, v.x);
    unsafeAtomicAdd(dst + 1, v.y);
    unsafeAtomicAdd(dst + 2, v.z);
    unsafeAtomicAdd(dst + 3, v.w);
    if (lane == 0) unsafeAtomicAdd(cnt + t, 1.0f);
}

// ---------------- mean + GEMM (WMMA f32 16x16x4) + bias + residual ----------------
// block = 256 threads = 8 waves. Block handles 16 node-rows x 128 output cols.
// Wave w handles the 16x16 output tile at columns n0 = 16*w.
#define LDS_STRIDE 130  // pad 128 -> 130 to break bank conflicts on row-strided reads
__global__ void wal_gemm_kernel(const float* __restrict__ agg,
                                const float* __restrict__ cnt,
                                const float* __restrict__ x,
                                const float* __restrict__ W,   // [128,128] row-major: W[j][k]
                                const float* __restrict__ bias,
                                float* __restrict__ out,
                                int N) {
    __shared__ float sA[16 * LDS_STRIDE];

    int m0 = blockIdx.x * 16;
    if (m0 >= N) return;
    bool full = (m0 + 16 <= N);    // true for every block when N % 16 == 0

    // Stage A tile: mean rows m0..m0+15 (sum/max(count,1)) into LDS.
    if (full) {
        for (int i = threadIdx.x; i < 16 * IN_DIM; i += blockDim.x) {
            int row = i >> 7;              // 0..15
            int col = i & (IN_DIM - 1);
            int gr  = m0 + row;
            float v = agg[(size_t)gr * IN_DIM + col];
            float c = cnt[gr];
            sA[row * LDS_STRIDE + col] = v / fmaxf(c, 1.0f);
        }
    } else {
        for (int i = threadIdx.x; i < 16 * IN_DIM; i += blockDim.x) {
            int row = i >> 7;
            int col = i & (IN_DIM - 1);
            int gr  = m0 + row;
            float v = 0.0f, c = 1.0f;
            if (gr < N) {
                v = agg[(size_t)gr * IN_DIM + col];
                c = cnt[gr];
            }
            sA[row * LDS_STRIDE + col] = v / fmaxf(c, 1.0f);
        }
    }
    __syncthreads();

    int wave  = threadIdx.x >> 5;      // 0..7
    int lane  = threadIdx.x & 31;
    int half  = lane >> 4;             // 0: K pair {0,1}, 1: K pair {2,3}
    int mlane = lane & 15;
    int n0    = wave * 16;

    v8f acc = {};
    const float2* Wp = (const float2*)(W + (size_t)(n0 + mlane) * IN_DIM);

    // K = 128 in 32 steps of 4, f32 WMMA keeps full reference precision.
#pragma unroll 8
    for (int k = 0; k < IN_DIM; k += 4) {
        int kk = k + 2 * half;
        v2f a;
        a.x = sA[mlane * LDS_STRIDE + kk];
        a.y = sA[mlane * LDS_STRIDE + kk + 1];
        float2 wv = Wp[kk >> 1];       // B[kk..kk+1][n0+mlane] = W[n0+mlane][kk..kk+1]
        v2f bf;
        bf.x = wv.x;
        bf.y = wv.y;
        acc = __builtin_amdgcn_wmma_f32_16x16x4_f32(
            /*neg_a=*/false, a, /*neg_b=*/false, bf,
            /*c_mod=*/(short)0, acc, /*reuse_a=*/false, /*reuse_b=*/false);
    }

    // Epilogue: out = x + (mean @ W^T) + b
    // acc[v] holds row (m0 + v + 8*half), col (n0 + mlane).
    int col = n0 + mlane;
    float bcol = bias[col];
    size_t base = (size_t)(m0 + 8 * half) * IN_DIM + col;   // row stride = IN_DIM floats
    const float* __restrict__ xb = x + base;
    float* __restrict__ ob = out + base;

    if (full) {
#pragma unroll
        for (int v = 0; v < 8; ++v) {
            ob[(size_t)v * IN_DIM] = acc[v] + xb[(size_t)v * IN_DIM] + bcol;
        }
    } else {
#pragma unroll
        for (int v = 0; v < 8; ++v) {
            int row = m0 + v + 8 * half;
            if (row < N) {
                ob[(size_t)v * IN_DIM] = acc[v] + xb[(size_t)v * IN_DIM] + bcol;
            }
        }
    }
}

extern "C" void kernel_launch(void* const* d_in, const int* in_sizes, int n_in,
                              void* d_out, int out_size, void* d_ws, size_t ws_size,
                              hipStream_t stream) {
    const float*     x    = (const float*)d_in[0];
    const long long* ei   = (const long long*)d_in[1];   // int64 [2, E]
    const float*     W    = (const float*)d_in[2];
    const float*     bias = (const float*)d_in[3];
    float*           out  = (float*)d_out;

    int N = in_sizes[0] / IN_DIM;   // 100000
    int E = in_sizes[1] / 2;        // 640000

    float* agg = (float*)d_ws;                    // [N,128] sums
    float* cnt = agg + (size_t)N * IN_DIM;        // [N] counts

    size_t ztot = (size_t)N * IN_DIM + (size_t)N;
    wal_zero_kernel<<<(unsigned)((ztot + 255) / 256), 256, 0, stream>>>(agg, ztot);

    // one wave32 per edge -> 8 edges per 256-thread block
    wal_scatter_kernel<<<(E + 7) / 8, 256, 0, stream>>>(x, ei, agg, cnt, E);

    wal_gemm_kernel<<<(N + 15) / 16, 256, 0, stream>>>(agg, cnt, x, W, bias, out, N);
}